// relative_depth_crit_27161373180109
// MI455X (gfx1250) — compile-verified
//
#include <hip/hip_runtime.h>

// Relative-depth ordinal loss, MI455X (gfx1250, wave32).
// Memory-bound gather+reduce: ~64 MB streamed index data + L2-resident gathers.
// Uses V_WMMA_F32_16X16X4_F32 with an all-ones A matrix as a 16-way parallel
// f32 accumulator (D[m][n] = sum_k B[k][n] + C[m][n]; with A==1 every row of D
// equals the column-sum vector, so sum(lane VGPR0) == 2 * total regardless of
// the exact B lane layout).

typedef float v2f __attribute__((ext_vector_type(2)));
typedef float v8f __attribute__((ext_vector_type(8)));

#define NBLK 1024
#define NTHR 256
#define PPTS 100000u   // P (points per batch); even, so an even element index
                       // and its +1 neighbor always share the same batch b.

__global__ __launch_bounds__(NTHR) void rdc_partial_kernel(
    const float* __restrict__ img,
    const int*   __restrict__ xA, const int* __restrict__ yA,
    const int*   __restrict__ xB, const int* __restrict__ yB,
    const int*   __restrict__ ord,
    float*       __restrict__ partials,
    unsigned nTotal)
{
    const unsigned g           = blockIdx.x * NTHR + threadIdx.x;
    const unsigned strideLanes = NBLK * NTHR;
    const unsigned nPairs      = nTotal >> 1;                       // nTotal even
    const unsigned iters       = (nPairs + strideLanes - 1u) / strideLanes; // uniform

    v2f ones; ones.x = 1.0f; ones.y = 1.0f;   // A-matrix (16x4) all ones
    v8f acc = {0.f,0.f,0.f,0.f,0.f,0.f,0.f,0.f};

    for (unsigned it = 0; it < iters; ++it) {
        unsigned pr    = g + it * strideLanes;        // pair index
        bool     valid = pr < nPairs;
        unsigned p0    = valid ? (pr << 1) : 0u;      // element index, 8B-aligned

        // Coalesced 8-byte non-temporal streams (read-once: keep image in L2).
        unsigned long long vxa = __builtin_nontemporal_load((const unsigned long long*)(xA  + p0));
        unsigned long long vya = __builtin_nontemporal_load((const unsigned long long*)(yA  + p0));
        unsigned long long vxb = __builtin_nontemporal_load((const unsigned long long*)(xB  + p0));
        unsigned long long vyb = __builtin_nontemporal_load((const unsigned long long*)(yB  + p0));
        unsigned long long vo  = __builtin_nontemporal_load((const unsigned long long*)(ord + p0));

        int xa0 = (int)(vxa & 0xffffffffull), xa1 = (int)(vxa >> 32);
        int ya0 = (int)(vya & 0xffffffffull), ya1 = (int)(vya >> 32);
        int xb0 = (int)(vxb & 0xffffffffull), xb1 = (int)(vxb >> 32);
        int yb0 = (int)(vyb & 0xffffffffull), yb1 = (int)(vyb >> 32);
        int o0  = (int)(vo  & 0xffffffffull), o1  = (int)(vo  >> 32);

        // Batch image base: H*W = 2^18, W = 2^9 -> pure shifts/adds.
        const float* im = img + ((size_t)(p0 / PPTS) << 18);

        float zA0 = im[(ya0 << 9) + xa0];
        float zB0 = im[(yb0 << 9) + xb0];
        float zA1 = im[(ya1 << 9) + xa1];
        float zB1 = im[(yb1 << 9) + xb1];

        float d0 = zA0 - zB0;
        float d1 = zA1 - zB1;
        float gt0 = (float)(o0 - 1);        // {-1, 0, 1}
        float gt1 = (float)(o1 - 1);

        // loss = |gt| * log1p(exp(-gt*d)) + (1-|gt|) * d*d  (branch-free selects)
        float sp0 = __logf(1.0f + __expf(-gt0 * d0));
        float sp1 = __logf(1.0f + __expf(-gt1 * d1));
        float l0  = (o0 == 1) ? (d0 * d0) : sp0;
        float l1  = (o1 == 1) ? (d1 * d1) : sp1;
        l0 = valid ? l0 : 0.0f;
        l1 = valid ? l1 : 0.0f;

        v2f bm; bm.x = l0; bm.y = l1;       // B-matrix (4x16): 64 values/wave

        // acc = ones(16x4) * bm(4x16) + acc : 16 parallel f32 column accumulators.
        // EXEC is all-ones here (no divergent branches in this loop).
        acc = __builtin_amdgcn_wmma_f32_16x16x4_f32(
            /*neg_a=*/false, ones, /*neg_b=*/false, bm,
            /*c_mod=*/(short)0, acc, /*reuse_a=*/false, /*reuse_b=*/false);
    }

    // acc[0] (VGPR0): lanes 0-15 = D[0][n], lanes 16-31 = D[8][n], rows identical
    // => sum over 32 lanes = 2 * (wave total).
    float wsum = acc[0];
    #pragma unroll
    for (int off = 16; off > 0; off >>= 1)
        wsum += __shfl_xor(wsum, off, 32);

    __shared__ float sdata[NTHR / 32];
    if ((threadIdx.x & 31) == 0)
        sdata[threadIdx.x >> 5] = wsum * 0.5f;
    __syncthreads();
    if (threadIdx.x == 0) {
        float s = 0.0f;
        #pragma unroll
        for (int w = 0; w < NTHR / 32; ++w) s += sdata[w];
        partials[blockIdx.x] = s;           // plain store: no init/atomics needed
    }
}

__global__ __launch_bounds__(256) void rdc_final_kernel(
    const float* __restrict__ partials, float* __restrict__ out,
    int nb, double invN)
{
    __shared__ double sd[256];
    double s = 0.0;
    for (int i = threadIdx.x; i < nb; i += 256)   // fixed order -> deterministic
        s += (double)partials[i];
    sd[threadIdx.x] = s;
    __syncthreads();
    #pragma unroll
    for (int off = 128; off > 0; off >>= 1) {
        if (threadIdx.x < off) sd[threadIdx.x] += sd[threadIdx.x + off];
        __syncthreads();
    }
    if (threadIdx.x == 0)
        out[0] = (float)(sd[0] * invN);
}

extern "C" void kernel_launch(void* const* d_in, const int* in_sizes, int n_in,
                              void* d_out, int out_size, void* d_ws, size_t ws_size,
                              hipStream_t stream)
{
    const float* img = (const float*)d_in[0];
    const int*   xA  = (const int*)d_in[1];   // jax x64 disabled -> int32
    const int*   yA  = (const int*)d_in[2];
    const int*   xB  = (const int*)d_in[3];
    const int*   yB  = (const int*)d_in[4];
    const int*   ord = (const int*)d_in[5];

    const unsigned n = (unsigned)in_sizes[1];     // B*P = 3,200,000
    float* partials  = (float*)d_ws;              // NBLK floats (4 KB)

    rdc_partial_kernel<<<NBLK, NTHR, 0, stream>>>(img, xA, yA, xB, yB, ord,
                                                  partials, n);
    rdc_final_kernel<<<1, 256, 0, stream>>>(partials, (float*)d_out,
                                            NBLK, 1.0 / (double)n);
}